// BDHScanner_4406636445723
// MI455X (gfx1250) — compile-verified
//
#include <hip/hip_runtime.h>
#include <hip/hip_bf16.h>
#include <math.h>

#define DIM    1024
#define SEQ    128
#define TOPK   51
#define DECAY  0.99f
#define INHIB  1.5f

typedef float v2f __attribute__((ext_vector_type(2)));
typedef float v8f __attribute__((ext_vector_type(8)));

// ---------------------------------------------------------------------------
// Kernel 1: sparsify all tokens in parallel (s_t depends only on x_t).
// One block (1024 threads) per token. Tie-correct top-k:
// keep pos[i] iff (#j: pos[j] > pos[i]) < K  <=>  pos[i] >= kth-largest.
// Also precompute alpha_t = 1 - c_t * INHIB.
// ---------------------------------------------------------------------------
__global__ __launch_bounds__(1024)
void bdh_sparsify(const float* __restrict__ x,
                  const float* __restrict__ cscore,
                  float* __restrict__ s_all,
                  float* __restrict__ alpha)
{
    __shared__ float pos[DIM];
    const int t = blockIdx.x;
    const int i = threadIdx.x;
    float v = x[(size_t)t * DIM + i];
    v = fmaxf(v, 0.0f);
    pos[i] = v;
    __syncthreads();
    int cnt = 0;
#pragma unroll 8
    for (int j = 0; j < DIM; ++j)
        cnt += (pos[j] > v) ? 1 : 0;       // LDS broadcast reads, no bank conflict
    const float s = (cnt < TOPK) ? v : 0.0f;
    s_all[(size_t)t * DIM + i] = s;
    if (i == 0) alpha[t] = 1.0f - cscore[t] * INHIB;
}

// ---------------------------------------------------------------------------
// Kernel 2 (per step): expectation e = s_t @ state_{t-1} via f32 WMMA GEVM.
// One wave per 16-column output tile (64 tiles -> 8 blocks x 8 waves).
// A (16x4 f32): row M=0 carries s; lanes with (lane&15)==0 hold the K-pair
//   (lane 0 -> K=0,1 ; lane 16 -> K=2,3), all other rows zero.
// B (4x16 f32): lane l, VGPR v holds B[K = v + 2*(l>>4)][N = l&15].
// ---------------------------------------------------------------------------
__global__ __launch_bounds__(256)
void bdh_expect(const float* __restrict__ prev,   // state_{t-1} (ignored if t0)
                const float* __restrict__ s,      // s_t [DIM]
                float* __restrict__ e,            // out [DIM]
                int t0)
{
    const int lane  = threadIdx.x & 31;
    const int wave  = blockIdx.x * (blockDim.x >> 5) + (threadIdx.x >> 5); // 0..63
    const int lcol  = lane & 15;
    const int khalf = lane >> 4;
    const int col   = wave * 16 + lcol;

    v8f acc = {};
    if (!t0) {
        for (int k = 0; k < DIM; k += 4) {
            v2f a;
            const float s0 = s[k + 2 * khalf];
            const float s1 = s[k + 2 * khalf + 1];
            a.x = (lcol == 0) ? s0 : 0.0f;    // only row M=0 populated
            a.y = (lcol == 0) ? s1 : 0.0f;
            v2f b;
            const float* m = prev + (size_t)(k + 2 * khalf) * DIM + col;
            b.x = m[0];
            b.y = m[DIM];
            acc = __builtin_amdgcn_wmma_f32_16x16x4_f32(
                false, a, false, b, (short)0, acc, false, false);
        }
    }
    // row M=0 of the result lives in VGPR0 on lanes 0..15
    if (lane < 16) e[col] = acc[0];
}

// ---------------------------------------------------------------------------
// Kernel 3 (per step): drift_t = || s_t - e || , single-block LDS reduction.
// ---------------------------------------------------------------------------
__global__ __launch_bounds__(1024)
void bdh_drift(const float* __restrict__ s,
               const float* __restrict__ e,
               float* __restrict__ out_drift)
{
    __shared__ float red[DIM];
    const int i = threadIdx.x;
    const float d = s[i] - e[i];
    red[i] = d * d;
    __syncthreads();
    for (int off = 512; off > 0; off >>= 1) {
        if (i < off) red[i] += red[i + off];
        __syncthreads();
    }
    if (i == 0) *out_drift = sqrtf(red[0]);
}

// ---------------------------------------------------------------------------
// Kernel 4 (per step): state_t = relu(state_{t-1}*DECAY + alpha * s s^T).
// One wave per 16x16 tile; the rank-1 outer product + accumulate is a single
// V_WMMA_F32_16X16X4_F32 with only K=0 populated and C = decayed prev tile.
// C/D layout: VGPR v, lane l -> row v + 8*(l>>4), col l&15 (coalesced rows).
// Writes directly into the d_out snapshot slot for step t.
// ---------------------------------------------------------------------------
__global__ __launch_bounds__(256)
void bdh_update(const float* __restrict__ prev,   // state_{t-1} (ignored if t0)
                const float* __restrict__ s,      // s_t
                const float* __restrict__ alpha_p,
                float* __restrict__ outstate,     // state_t (d_out slice)
                int t, int t0)
{
    const int lane = threadIdx.x & 31;
    const int wave = blockIdx.x * (blockDim.x >> 5) + (threadIdx.x >> 5); // 0..4095
    const int tm = wave >> 6;                 // 16-row tile index
    const int tn = wave & 63;                 // 16-col tile index
    const int lcol = lane & 15;
    const int half = lane >> 4;
    const int rowbase = tm * 16;
    const int col = tn * 16 + lcol;
    const float ascale = alpha_p[t];

    v8f c = {};
    if (!t0) {
#pragma unroll
        for (int v = 0; v < 8; ++v)
            c[v] = prev[(size_t)(rowbase + v + 8 * half) * DIM + col] * DECAY;
    }
    v2f a;                                     // A[:,0] = s_rows * alpha
    a.x = (lane < 16) ? s[rowbase + lane] * ascale : 0.0f;
    a.y = 0.0f;
    v2f b;                                     // B[0,:] = s_cols
    b.x = (lane < 16) ? s[tn * 16 + lane] : 0.0f;
    b.y = 0.0f;

    v8f d = __builtin_amdgcn_wmma_f32_16x16x4_f32(
        false, a, false, b, (short)0, c, false, false);

#pragma unroll
    for (int v = 0; v < 8; ++v)
        outstate[(size_t)(rowbase + v + 8 * half) * DIM + col] = fmaxf(d[v], 0.0f);
}

// ---------------------------------------------------------------------------
extern "C" void kernel_launch(void* const* d_in, const int* in_sizes, int n_in,
                              void* d_out, int out_size, void* d_ws, size_t ws_size,
                              hipStream_t stream)
{
    const float* emb = (const float*)d_in[0];   // [S, D] f32
    const float* cs  = (const float*)d_in[1];   // [S]    f32

    float* out        = (float*)d_out;
    float* out_drift  = out;                    // [S]
    float* out_states = out + SEQ;              // [S, D, D]

    float* ws       = (float*)d_ws;
    float* ws_s     = ws;                       // S*D floats
    float* ws_alpha = ws + (size_t)SEQ * DIM;   // S floats
    float* ws_e     = ws_alpha + SEQ;           // D floats

    // Parallel precompute of all sparsified vectors + alphas.
    bdh_sparsify<<<SEQ, 1024, 0, stream>>>(emb, cs, ws_s, ws_alpha);

    // Sequential scan; state_t lives directly in its output snapshot slot.
    for (int t = 0; t < SEQ; ++t) {
        const int t0 = (t == 0) ? 1 : 0;
        const float* prev = t0 ? out_states
                               : out_states + (size_t)(t - 1) * DIM * DIM;
        const float* st = ws_s + (size_t)t * DIM;

        bdh_expect<<<8, 256, 0, stream>>>(prev, st, ws_e, t0);
        bdh_drift <<<1, 1024, 0, stream>>>(st, ws_e, out_drift + t);
        bdh_update<<<512, 256, 0, stream>>>(prev, st, ws_alpha,
                                            out_states + (size_t)t * DIM * DIM,
                                            t, t0);
    }
}